// FNO_multi_16758962389660
// MI455X (gfx1250) — compile-verified
//
#include <hip/hip_runtime.h>
#include <cmath>

// ---------------------------------------------------------------------------
// FNO (B=4,V=2,S=256,W=32,M=16,NB=6) for gfx1250 (MI455X), wave32 + WMMA f32.
// Spectral conv implemented as truncated-DFT GEMM chain (M=16 << S=256), so
// every heavy op is a V_WMMA_F32_16X16X4_F32 GEMM. fp32 throughout: all GEMMs
// are HBM-bound (AI ~8-16 flop/B at 23.3 TB/s), so f32 WMMA is precision-free.
// Streaming GEMMs stage their DFT basis into LDS via the Tensor Data Mover
// (TENSOR_LOAD_TO_LDS + s_wait_tensorcnt), and each wave computes a 16x32
// strip so every A fragment feeds two WMMAs (halves the HBM A-stream).
// ---------------------------------------------------------------------------

typedef float v2f __attribute__((ext_vector_type(2)));
typedef float v8f __attribute__((ext_vector_type(8)));
typedef unsigned int u32x4 __attribute__((ext_vector_type(4)));
typedef int i32x4 __attribute__((ext_vector_type(4)));
typedef int i32x8 __attribute__((ext_vector_type(8)));

#define SDIM 256
#define SS   65536            // SDIM*SDIM
#define NW   32               // channels W
#define NBV  8                // B*V
#define NPL  256              // NBV*NW planes
#define V8ZERO {0.f,0.f,0.f,0.f,0.f,0.f,0.f,0.f}

__device__ __forceinline__ v8f wmma4(v2f a, v2f b, v8f c) {
  // D(16x16,f32) = A(16x4,f32) x B(4x16,f32) + C
  return __builtin_amdgcn_wmma_f32_16x16x4_f32(false, a, false, b, (short)0, c, false, false);
}

__device__ __forceinline__ float gelu_f(float v) {
  return 0.5f * v * (1.0f + erff(v * 0.70710678118654752440f));
}

// ---------------------------------------------------------------------------
// TDM: 1-D copy of nElems f32 from global to LDS (D# per CDNA5 ISA ch.8).
// group0: count=1 | lds_addr | global_addr[56:0] | type=2("image")
// group1: data_size=4B, tensor_dim0=tile_dim0=nElems, tensor_dim1=1
// Issued once per workgroup (wave 0); completion via s_wait_tensorcnt 0.
// This toolchain exposes the 6-arg builtin (g0, g1, g2, g3, g4, cpol).
// ---------------------------------------------------------------------------
__device__ __forceinline__ void tdm_load_1d(const float* gsrc, unsigned lds_off, int nElems) {
#if __has_builtin(__builtin_amdgcn_tensor_load_to_lds)
  unsigned long long ga = (unsigned long long)(uintptr_t)gsrc;
  u32x4 g0 = {0u, 0u, 0u, 0u};
  g0[0] = 1u;                                                  // count=1, is_restore=0
  g0[1] = lds_off;                                             // lds_addr (bytes)
  g0[2] = (unsigned)(ga & 0xFFFFFFFFu);                        // global_addr[31:0]
  g0[3] = (unsigned)((ga >> 32) & 0x01FFFFFFu) | (2u << 30);   // global_addr[56:32], type=2
  i32x8 g1 = {0, 0, 0, 0, 0, 0, 0, 0};
  g1[0] = (2 << 16);                                           // wg_mask=0, data_size=2 (4B)
  g1[1] = (nElems & 0xFFFF) << 16;                             // tensor_dim0[15:0]
  g1[2] = ((nElems >> 16) & 0xFFFF) | (1 << 16);               // tensor_dim0[31:16], tensor_dim1=1
  g1[3] = (nElems & 0xFFFF) << 16;                             // tile_dim0
  g1[4] = 0;                                                   // tile_dim1=0, tile_dim2=0
  g1[5] = nElems;                                              // tensor_dim0_stride[31:0]
  g1[6] = 0;
  g1[7] = 0;
  i32x4 z4 = {0, 0, 0, 0};
  i32x8 z8 = {0, 0, 0, 0, 0, 0, 0, 0};
  __builtin_amdgcn_tensor_load_to_lds(g0, g1, z4, z4, z8, 0);
  __builtin_amdgcn_s_wait_tensorcnt(0);
#else
  (void)gsrc; (void)lds_off; (void)nElems;
#endif
}

// ---------------------------------------------------------------------------
// Basis tables (in workspace):
//  Bx  [256][32] : fwd-x DFT   cols 0..15 = cos, 16..31 = -sin   (e^{-i})
//  Ay  [64][256] : fwd-y DFT   rows 0..31 = cos, 32..63 = sin,  ky in {0..15,-16..-1}
//  A3a [256][64] : inv-y (gr)  cols 0..31 = cos, 32..63 = -sin  (e^{+i})
//  A3b [256][64] : inv-y (gi)  cols 0..31 = sin, 32..63 = cos
//  Bix [32][256] : inv-x       rows 0..15 = a*cos/S^2, 16..31 = -a*sin/S^2 (a: herm x2)
// ---------------------------------------------------------------------------
__global__ void k_basis(float* __restrict__ basis) {
  int i = blockIdx.x * 256 + threadIdx.x;     // 65536 total
  const float step = 6.28318530717958647692f / 256.0f;
  float val = 0.0f;
  if (i < 8192) {                              // Bx
    int x = i >> 5, j = i & 31, jj = j & 15;
    float th = ((jj * x) & 255) * step;
    val = (j < 16) ? cosf(th) : -sinf(th);
  } else if (i < 24576) {                      // Ay
    int e = i - 8192, r = e >> 8, y = e & 255;
    int rr = r & 31; int kk = (rr < 16) ? rr : rr - 32;
    int q = ((kk * y) % 256 + 256) & 255;
    float th = q * step;
    val = (r < 32) ? cosf(th) : sinf(th);
  } else if (i < 40960) {                      // A3a
    int e = i - 24576, y = e >> 6, k = e & 63;
    int km = k & 31; int kk = (km < 16) ? km : km - 32;
    int q = ((kk * y) % 256 + 256) & 255;
    float th = q * step;
    val = (k < 32) ? cosf(th) : -sinf(th);
  } else if (i < 57344) {                      // A3b
    int e = i - 40960, y = e >> 6, k = e & 63;
    int km = k & 31; int kk = (km < 16) ? km : km - 32;
    int q = ((kk * y) % 256 + 256) & 255;
    float th = q * step;
    val = (k < 32) ? sinf(th) : cosf(th);
  } else {                                     // Bix
    int e = i - 57344, j = e >> 8, x = e & 255, jj = j & 15;
    float amp = ((jj == 0) ? 1.0f : 2.0f) * (1.0f / 65536.0f);
    float th = ((jj * x) & 255) * step;
    val = (j < 16) ? amp * cosf(th) : -amp * sinf(th);
  }
  basis[i] = val;
}

// ---------------------------------------------------------------------------
// fc0: h[bc][p] = fc0_b[w] + sum_k in[k]*fc0_w[k][w], in = [x(10), gx, gy]
// planes layout: plane bc = (b*V+v)*32 + w, contiguous 256x256 image.
// ---------------------------------------------------------------------------
__global__ __launch_bounds__(256) void k_fc0(const float* __restrict__ xin,
                                             const float* __restrict__ w,
                                             const float* __restrict__ bvec,
                                             float* __restrict__ H) {
  __shared__ float Xs[64][10];
  int blk = blockIdx.x;                        // 8192 = NBV * 1024
  int bv = blk >> 10, p0 = (blk & 1023) << 6;  // 64 points per block
  int t = threadIdx.x;
  for (int e = t; e < 640; e += 256) {
    int pt = e / 10, k = e % 10;
    Xs[pt][k] = xin[((size_t)bv * SS + p0 + pt) * 10 + k];
  }
  __syncthreads();
  #pragma unroll
  for (int r = 0; r < 8; r++) {
    int idx = t + (r << 8);                    // 0..2047 = 32ch x 64pt
    int wch = idx >> 6, pt = idx & 63;
    int p = p0 + pt, yy = p >> 8, xx = p & 255;
    float gxv = 9.5f + yy * (1.0f / 255.0f);
    float gyv = -0.5f + xx * (1.0f / 255.0f);
    float acc = bvec[wch];
    #pragma unroll
    for (int k = 0; k < 10; k++) acc += Xs[pt][k] * w[k * 32 + wch];
    acc += gxv * w[320 + wch] + gyv * w[352 + wch];
    H[((size_t)bv * 32 + wch) * SS + p0 + pt] = acc;
  }
}

// ---------------------------------------------------------------------------
// fwd-x: per plane G(256x32) = H(256x256) @ Bx(256x32).
// Basis in LDS via TDM; 1 wave = 16x32 strip (A fragment reused for 2 WMMAs).
// ---------------------------------------------------------------------------
__global__ __launch_bounds__(128) void k_fwdx(const float* __restrict__ H,
                                              const float* __restrict__ Bxm,
                                              float* __restrict__ G) {
  __shared__ float Bs[8192];                           // Bx 256x32
  int wid = threadIdx.x >> 5, lane = threadIdx.x & 31;
#if __has_builtin(__builtin_amdgcn_tensor_load_to_lds)
  if (wid == 0) tdm_load_1d(Bxm, (unsigned)(uintptr_t)(void*)Bs, 8192);
#else
  for (int e = threadIdx.x; e < 8192; e += 128) Bs[e] = Bxm[e];
#endif
  __syncthreads();
  int wtile = blockIdx.x * 4 + wid;                    // 4096 tiles (pl, yt)
  int yt = wtile & 15, pl = wtile >> 4;
  int m = lane & 15, kp = (lane >> 4) << 1;
  const float* A = H + (size_t)pl * SS + (size_t)(yt * 16 + m) * SDIM + kp;
  const float* B0 = &Bs[kp * 32 + m];
  const float* B1 = &Bs[kp * 32 + 16 + m];
  v8f acc0 = V8ZERO, acc1 = V8ZERO;
  for (int k = 0; k < 256; k += 4) {
    if ((k & 63) == 0) __builtin_prefetch(A + k + 128, 0, 1);
    v2f a;  a.x  = A[k];        a.y  = A[k + 1];
    v2f b0; b0.x = B0[k * 32];  b0.y = B0[k * 32 + 32];
    v2f b1; b1.x = B1[k * 32];  b1.y = B1[k * 32 + 32];
    acc0 = wmma4(a, b0, acc0);
    acc1 = wmma4(a, b1, acc1);
  }
  float* Gp = G + (size_t)pl * 8192 + (size_t)(yt * 16) * 32 + m;
  int m0 = (lane >> 4) << 3;
  #pragma unroll
  for (int i = 0; i < 8; i++) {
    Gp[(size_t)(m0 + i) * 32]      = acc0[i];
    Gp[(size_t)(m0 + i) * 32 + 16] = acc1[i];
  }
}

// ---------------------------------------------------------------------------
// fwd-y: per plane P(64x32) = Ay(64x256) @ G(256x32)
// ---------------------------------------------------------------------------
__global__ __launch_bounds__(128) void k_fwdy(const float* __restrict__ G,
                                              const float* __restrict__ Aym,
                                              float* __restrict__ P) {
  int lane = threadIdx.x & 31;
  int wtile = blockIdx.x * 4 + (threadIdx.x >> 5);     // 2048 tiles
  int nt = wtile & 1, rt = (wtile >> 1) & 3, pl = wtile >> 3;
  int m = lane & 15, kp = (lane >> 4) << 1;
  const float* A  = Aym + (size_t)(rt * 16 + m) * 256 + kp;
  const float* Bp = G + (size_t)pl * 8192 + (size_t)kp * 32 + nt * 16 + m;
  v8f acc = V8ZERO;
  for (int k = 0; k < 256; k += 4) {
    v2f a; a.x = A[k]; a.y = A[k + 1];
    v2f b; b.x = Bp[k * 32]; b.y = Bp[k * 32 + 32];
    acc = wmma4(a, b, acc);
  }
  float* Pp = P + (size_t)pl * 2048 + (size_t)(rt * 16) * 32 + nt * 16 + m;
  int m0 = (lane >> 4) << 3;
  #pragma unroll
  for (int i = 0; i < 8; i++) Pp[(size_t)(m0 + i) * 32] = acc[i];
}

// ---------------------------------------------------------------------------
// mode mix: per (v,ki,kx): Y[b,o] = sum_i Wc[i,o] * X[b,i]  (complex)
//  X from P: Xr = Pcos@Gr + Psin@Gi ; Xi = Pcos@Gi - Psin@Gr
//  ki<16 -> w1 (ky=0..15); ki>=16 -> w2 (ky=240..255)
// ---------------------------------------------------------------------------
__global__ __launch_bounds__(128) void k_modemix(const float* __restrict__ P, float* __restrict__ Y,
                                                 const float* __restrict__ w1r, const float* __restrict__ w1i,
                                                 const float* __restrict__ w2r, const float* __restrict__ w2i) {
  __shared__ float Wr[32][32], Wi[32][32];
  __shared__ float Xr[4][32], Xi[4][32];
  int idx = blockIdx.x;                        // 1024 = V*32*16
  int kx = idx & 15, ki = (idx >> 4) & 31, v = idx >> 9;
  int r = ki & 15;
  const float* wr = (ki < 16) ? w1r : w2r;
  const float* wi = (ki < 16) ? w1i : w2i;
  int t = threadIdx.x;
  for (int e = t; e < 1024; e += 128) {
    int ic = e >> 5, oc = e & 31;
    size_t wix = ((size_t)((ic * 32 + oc) * 2 + v) * 16 + r) * 16 + kx;
    Wr[ic][oc] = wr[wix];
    Wi[ic][oc] = wi[wix];
  }
  {
    int b = t >> 5, ic = t & 31;
    const float* Pp = P + (size_t)((b * 2 + v) * 32 + ic) * 2048;
    Xr[b][ic] = Pp[ki * 32 + kx]      + Pp[(32 + ki) * 32 + 16 + kx];
    Xi[b][ic] = Pp[ki * 32 + 16 + kx] - Pp[(32 + ki) * 32 + kx];
  }
  __syncthreads();
  int b = t >> 5, oc = t & 31;
  float ar = 0.f, ai = 0.f;
  #pragma unroll
  for (int ic = 0; ic < 32; ic++) {
    float wrv = Wr[ic][oc], wiv = Wi[ic][oc];
    float xr = Xr[b][ic], xi = Xi[b][ic];
    ar += wrv * xr - wiv * xi;
    ai += wrv * xi + wiv * xr;
  }
  float* Yp = Y + (size_t)((b * 2 + v) * 32 + oc) * 1024;   // Ystack [64][16]
  Yp[ki * 16 + kx]        = ar;
  Yp[(32 + ki) * 16 + kx] = ai;
}

// ---------------------------------------------------------------------------
// inv-y: per plane, half hf: Gy[:, hf*16+0..15] = A3{a,b}(256x64) @ Ystack(64x16)
// ---------------------------------------------------------------------------
__global__ __launch_bounds__(128) void k_invy(const float* __restrict__ Y,
                                              const float* __restrict__ A3a,
                                              const float* __restrict__ A3b,
                                              float* __restrict__ Gy) {
  int lane = threadIdx.x & 31;
  int wtile = blockIdx.x * 4 + (threadIdx.x >> 5);     // 8192 tiles
  int hf = wtile & 1, yt = (wtile >> 1) & 15, pl = wtile >> 5;
  int m = lane & 15, kp = (lane >> 4) << 1;
  const float* A  = (hf ? A3b : A3a) + (size_t)(yt * 16 + m) * 64 + kp;
  const float* Bp = Y + (size_t)pl * 1024 + (size_t)kp * 16 + m;
  v8f acc = V8ZERO;
  #pragma unroll
  for (int k = 0; k < 64; k += 4) {
    v2f a; a.x = A[k]; a.y = A[k + 1];
    v2f b; b.x = Bp[k * 16]; b.y = Bp[k * 16 + 16];
    acc = wmma4(a, b, acc);
  }
  float* Gp = Gy + (size_t)pl * 8192 + (size_t)(yt * 16) * 32 + hf * 16 + m;
  int m0 = (lane >> 4) << 3;
  #pragma unroll
  for (int i = 0; i < 8; i++) Gp[(size_t)(m0 + i) * 32] = acc[i];
}

// ---------------------------------------------------------------------------
// inv-x: per plane H1(256x256) = Gy(256x32) @ Bix(32x256).
// Basis in LDS via TDM; 1 wave = 16x32 strip (A fragment reused for 2 WMMAs).
// ---------------------------------------------------------------------------
__global__ __launch_bounds__(128) void k_invx(const float* __restrict__ Gy,
                                              const float* __restrict__ Bixm,
                                              float* __restrict__ H1) {
  __shared__ float Bs[8192];                           // Bix 32x256
  int wid = threadIdx.x >> 5, lane = threadIdx.x & 31;
#if __has_builtin(__builtin_amdgcn_tensor_load_to_lds)
  if (wid == 0) tdm_load_1d(Bixm, (unsigned)(uintptr_t)(void*)Bs, 8192);
#else
  for (int e = threadIdx.x; e < 8192; e += 128) Bs[e] = Bixm[e];
#endif
  __syncthreads();
  int wtile = blockIdx.x * 4 + wid;                    // 32768 strips
  int xh = wtile & 7, yt = (wtile >> 3) & 15, pl = wtile >> 7;
  int m = lane & 15, kp = (lane >> 4) << 1;
  const float* A  = Gy + (size_t)pl * 8192 + (size_t)(yt * 16 + m) * 32 + kp;
  const float* B0 = &Bs[kp * 256 + xh * 32 + m];
  const float* B1 = B0 + 16;
  v8f acc0 = V8ZERO, acc1 = V8ZERO;
  #pragma unroll
  for (int k = 0; k < 32; k += 4) {
    v2f a;  a.x  = A[k];         a.y  = A[k + 1];
    v2f b0; b0.x = B0[k * 256];  b0.y = B0[k * 256 + 256];
    v2f b1; b1.x = B1[k * 256];  b1.y = B1[k * 256 + 256];
    acc0 = wmma4(a, b0, acc0);
    acc1 = wmma4(a, b1, acc1);
  }
  float* Hp = H1 + (size_t)pl * SS + (size_t)(yt * 16) * 256 + xh * 32 + m;
  int m0 = (lane >> 4) << 3;
  #pragma unroll
  for (int i = 0; i < 8; i++) {
    Hp[(size_t)(m0 + i) * 256]      = acc0[i];
    Hp[(size_t)(m0 + i) * 256 + 16] = acc1[i];
  }
}

// ---------------------------------------------------------------------------
// pointwise: out = gelu( mlp2@gelu(mlp1@hspec+b1)+b2 + ww@h+wb + bw@grid+bb ) [+resid]
// 1 wave = 16 points x 32 channels, all three GEMMs via WMMA through LDS tiles.
// Writes in-place over hspec (tile read fully staged before writeback).
// ---------------------------------------------------------------------------
__global__ __launch_bounds__(128) void k_pointwise(float* hspec, const float* __restrict__ hin,
    const float* resid,
    const float* __restrict__ m1w, const float* __restrict__ m1b,
    const float* __restrict__ m2w, const float* __restrict__ m2b,
    const float* __restrict__ wwm, const float* __restrict__ wwb,
    const float* __restrict__ bwm, const float* __restrict__ bbm) {
  __shared__ float Ts[4][32][17];
  __shared__ float Th[4][32][17];
  __shared__ float Tg[4][32][17];
  int wid = threadIdx.x >> 5, lane = threadIdx.x & 31;
  int tile = blockIdx.x * 4 + wid;             // 32768 tiles
  int bv = tile >> 12, p0 = (tile & 4095) << 4;
  size_t base = (size_t)bv * 32 * SS + p0;
  {
    const float* sp = hspec + base + (size_t)lane * SS;
    const float* hp = hin   + base + (size_t)lane * SS;
    #pragma unroll
    for (int j = 0; j < 16; j++) { Ts[wid][lane][j] = sp[j]; Th[wid][lane][j] = hp[j]; }
  }
  __syncthreads();
  int n = lane & 15, kp = (lane >> 4) << 1, m0 = (lane >> 4) << 3;
  // stage 1: Tg = gelu(mlp1 @ Ts + b1)
  #pragma unroll
  for (int mt = 0; mt < 2; mt++) {
    v8f acc = V8ZERO;
    const float* Aw = m1w + (size_t)(mt * 16 + n) * 32 + kp;
    #pragma unroll
    for (int k = 0; k < 32; k += 4) {
      v2f a; a.x = Aw[k]; a.y = Aw[k + 1];
      v2f b; b.x = Ts[wid][k + kp][n]; b.y = Ts[wid][k + kp + 1][n];
      acc = wmma4(a, b, acc);
    }
    #pragma unroll
    for (int i = 0; i < 8; i++) {
      int o = mt * 16 + m0 + i;
      Tg[wid][o][n] = gelu_f(acc[i] + m1b[o]);
    }
  }
  __syncthreads();
  // stage 2: mlp2@Tg + ww@Th, epilogue
  int p = p0 + n, yy = p >> 8, xx = p & 255;
  float gxv = 9.5f + yy * (1.0f / 255.0f);
  float gyv = -0.5f + xx * (1.0f / 255.0f);
  #pragma unroll
  for (int mt = 0; mt < 2; mt++) {
    v8f acc = V8ZERO;
    const float* A2 = m2w + (size_t)(mt * 16 + n) * 32 + kp;
    const float* A3 = wwm + (size_t)(mt * 16 + n) * 32 + kp;
    #pragma unroll
    for (int k = 0; k < 32; k += 4) {
      v2f a; a.x = A2[k]; a.y = A2[k + 1];
      v2f b; b.x = Tg[wid][k + kp][n]; b.y = Tg[wid][k + kp + 1][n];
      acc = wmma4(a, b, acc);
    }
    #pragma unroll
    for (int k = 0; k < 32; k += 4) {
      v2f a; a.x = A3[k]; a.y = A3[k + 1];
      v2f b; b.x = Th[wid][k + kp][n]; b.y = Th[wid][k + kp + 1][n];
      acc = wmma4(a, b, acc);
    }
    #pragma unroll
    for (int i = 0; i < 8; i++) {
      int o = mt * 16 + m0 + i;
      float vo = acc[i] + m2b[o] + wwb[o] + bwm[2 * o] * gxv + bwm[2 * o + 1] * gyv + bbm[o];
      vo = gelu_f(vo);
      size_t oi = base + (size_t)o * SS + n;
      if (resid) vo += resid[oi];
      hspec[oi] = vo;
    }
  }
}

// ---------------------------------------------------------------------------
// head: out = gelu(h @ fc1 + b1) @ fc2 + b2, hidden 256 done in 64-wide chunks.
// ---------------------------------------------------------------------------
__global__ __launch_bounds__(128) void k_head(const float* __restrict__ H,
                                              const float* __restrict__ f1w, const float* __restrict__ f1b,
                                              const float* __restrict__ f2w, const float* __restrict__ f2b,
                                              float* __restrict__ out) {
  __shared__ float Th[4][32][17];
  __shared__ float Hc[4][64][17];
  int wid = threadIdx.x >> 5, lane = threadIdx.x & 31;
  int tile = blockIdx.x * 4 + wid;             // 32768 tiles
  int bv = tile >> 12, p0 = (tile & 4095) << 4;
  size_t base = (size_t)bv * 32 * SS + p0;
  {
    const float* hp = H + base + (size_t)lane * SS;
    #pragma unroll
    for (int j = 0; j < 16; j++) Th[wid][lane][j] = hp[j];
  }
  __syncthreads();
  int n = lane & 15, kp = (lane >> 4) << 1, m0 = (lane >> 4) << 3, g = lane >> 4;
  float acc5[5] = {0.f, 0.f, 0.f, 0.f, 0.f};
  for (int ci = 0; ci < 4; ci++) {
    int j0 = ci * 64;
    #pragma unroll
    for (int mt = 0; mt < 4; mt++) {
      v8f acc = V8ZERO;
      const float* Aw = f1w + (size_t)kp * 256 + (j0 + mt * 16 + n);  // A[m][k]=f1w[k*256+j]
      #pragma unroll
      for (int k = 0; k < 32; k += 4) {
        v2f a; a.x = Aw[(size_t)k * 256]; a.y = Aw[(size_t)(k + 1) * 256];
        v2f b; b.x = Th[wid][k + kp][n]; b.y = Th[wid][k + kp + 1][n];
        acc = wmma4(a, b, acc);
      }
      #pragma unroll
      for (int i = 0; i < 8; i++) {
        int jl = mt * 16 + m0 + i;
        Hc[wid][jl][n] = gelu_f(acc[i] + f1b[j0 + jl]);
      }
    }
    __syncthreads();
    for (int tt = 0; tt < 32; tt++) {
      int jl = 2 * tt + g;
      float hv = Hc[wid][jl][n];
      const float* wrow = f2w + (size_t)(j0 + jl) * 5;
      #pragma unroll
      for (int s = 0; s < 5; s++) acc5[s] += hv * wrow[s];
    }
    __syncthreads();
  }
  #pragma unroll
  for (int s = 0; s < 5; s++) acc5[s] += __shfl_xor(acc5[s], 16, 32);
  if (lane < 16) {
    size_t op = ((size_t)bv * SS + p0 + n) * 5;
    #pragma unroll
    for (int s = 0; s < 5; s++) out[op + s] = acc5[s] + f2b[s];
  }
}

// ---------------------------------------------------------------------------
// Host orchestration
// ---------------------------------------------------------------------------
extern "C" void kernel_launch(void* const* d_in, const int* in_sizes, int n_in,
                              void* d_out, int out_size, void* d_ws, size_t ws_size,
                              hipStream_t stream) {
  (void)in_sizes; (void)n_in; (void)out_size; (void)ws_size;
  const float* x      = (const float*)d_in[0];
  const float* fc0_w  = (const float*)d_in[1];
  const float* fc0_b  = (const float*)d_in[2];
  const float* w1r    = (const float*)d_in[3];
  const float* w1i    = (const float*)d_in[4];
  const float* w2r    = (const float*)d_in[5];
  const float* w2i    = (const float*)d_in[6];
  const float* mlp1_w = (const float*)d_in[7];
  const float* mlp1_b = (const float*)d_in[8];
  const float* mlp2_w = (const float*)d_in[9];
  const float* mlp2_b = (const float*)d_in[10];
  const float* ww     = (const float*)d_in[11];
  const float* wb     = (const float*)d_in[12];
  const float* bw     = (const float*)d_in[13];
  const float* bb     = (const float*)d_in[14];
  const float* fc1_w  = (const float*)d_in[15];
  const float* fc1_b  = (const float*)d_in[16];
  const float* fc2_w  = (const float*)d_in[17];
  const float* fc2_b  = (const float*)d_in[18];

  float* wsf  = (float*)d_ws;
  float* basis = wsf;                             // 65536
  float* Bx  = basis;
  float* Ay  = basis + 8192;
  float* A3a = basis + 24576;
  float* A3b = basis + 40960;
  float* Bix = basis + 57344;
  float* G    = wsf + 65536;                      // NPL*256*32
  float* Pm   = G   + (size_t)NPL * 8192;         // NPL*64*32
  float* Ym   = Pm  + (size_t)NPL * 2048;         // NPL*64*16
  float* Gy   = Ym  + (size_t)NPL * 1024;         // NPL*256*32
  float* bufA = Gy  + (size_t)NPL * 8192;         // NPL*SS each
  float* bufB = bufA + (size_t)NPL * SS;
  float* bufC = bufB + (size_t)NPL * SS;

  k_basis<<<256, 256, 0, stream>>>(basis);
  k_fc0<<<8192, 256, 0, stream>>>(x, fc0_w, fc0_b, bufA);

  const size_t WSPEC = 524288;   // W*W*V*M*M per block
  auto spectral = [&](int i, const float* hin, float* hspec) {
    k_fwdx<<<1024, 128, 0, stream>>>(hin, Bx, G);
    k_fwdy<<<512, 128, 0, stream>>>(G, Ay, Pm);
    k_modemix<<<1024, 128, 0, stream>>>(Pm, Ym, w1r + i * WSPEC, w1i + i * WSPEC,
                                        w2r + i * WSPEC, w2i + i * WSPEC);
    k_invy<<<2048, 128, 0, stream>>>(Ym, A3a, A3b, Gy);
    k_invx<<<8192, 128, 0, stream>>>(Gy, Bix, hspec);
  };
  auto pointwise = [&](int i, float* hspec, const float* hin, const float* resid) {
    k_pointwise<<<8192, 128, 0, stream>>>(hspec, hin, resid,
        mlp1_w + i * 1024, mlp1_b + i * 32, mlp2_w + i * 1024, mlp2_b + i * 32,
        ww + i * 1024, wb + i * 32, bw + i * 64, bb + i * 32);
  };

  // x0 = block0(h); h = block1(x0); h = block2(h)+x0;
  // x1s = block3(h); h = block4(x1s); h = block5(h)+x1s
  spectral(0, bufA, bufB); pointwise(0, bufB, bufA, nullptr);   // x0 in B
  spectral(1, bufB, bufC); pointwise(1, bufC, bufB, nullptr);   // h in C
  spectral(2, bufC, bufA); pointwise(2, bufA, bufC, bufB);      // h in A (+x0)
  spectral(3, bufA, bufB); pointwise(3, bufB, bufA, nullptr);   // x1s in B
  spectral(4, bufB, bufC); pointwise(4, bufC, bufB, nullptr);   // h in C
  spectral(5, bufC, bufA); pointwise(5, bufA, bufC, bufB);      // h in A (+x1s)

  k_head<<<8192, 128, 0, stream>>>(bufA, fc1_w, fc1_b, fc2_w, fc2_b, (float*)d_out);
}